// GCN_558345748541
// MI455X (gfx1250) — compile-verified
//
#include <hip/hip_runtime.h>

#define D_FEAT 128
#define N_CLASSES 4

typedef __attribute__((ext_vector_type(2))) float v2f;
typedef __attribute__((ext_vector_type(8))) float v8f;

// ---------------------------------------------------------------------------
// Utility: zero a buffer (float4 granularity; all buffers are 16B multiples)
// ---------------------------------------------------------------------------
__global__ __launch_bounds__(256) void zero_kernel(float4* __restrict__ p, long long n4) {
  long long i = (long long)blockIdx.x * blockDim.x + threadIdx.x;
  if (i < n4) p[i] = make_float4(0.f, 0.f, 0.f, 0.f);
}

// ---------------------------------------------------------------------------
// deg[dst[e]] += ew[e]   (self-loop "+1" folded into dinv_kernel)
// ---------------------------------------------------------------------------
__global__ __launch_bounds__(256) void deg_accum_kernel(const int* __restrict__ dst,
                                                        const float* __restrict__ ew,
                                                        float* __restrict__ deg, int e) {
  int i = blockIdx.x * blockDim.x + threadIdx.x;
  if (i < e) atomicAdd(&deg[dst[i]], ew[i]);
}

// deg -> rsqrt(deg + 1)  (deg+1 >= 1 always, matches reference's where(deg>0))
__global__ __launch_bounds__(256) void dinv_kernel(float* __restrict__ deg, int n) {
  int i = blockIdx.x * blockDim.x + threadIdx.x;
  if (i < n) deg[i] = rsqrtf(deg[i] + 1.0f);
}

// norm[e] = dinv[src[e]] * ew[e] * dinv[dst[e]]
__global__ __launch_bounds__(256) void norm_kernel(const int* __restrict__ src,
                                                   const int* __restrict__ dst,
                                                   const float* __restrict__ ew,
                                                   const float* __restrict__ dinv,
                                                   float* __restrict__ norm, int e) {
  int i = blockIdx.x * blockDim.x + threadIdx.x;
  if (i < e) norm[i] = dinv[src[i]] * ew[i] * dinv[dst[i]];
}

// ---------------------------------------------------------------------------
// H[n x 128] = X[n x 128] @ W[128 x 128], fp32, CDNA5 V_WMMA_F32_16X16X4_F32.
// One wave (32 lanes) produces one 16x16 tile of H; 32 chained WMMAs over K.
// A layout (16x4 f32): lane = row (lane&15); VGPR j holds K = j + 2*(lane>>4)
// B layout (4x16 f32): lane = col (lane&15); VGPR j holds K = j + 2*(lane>>4)
// C/D layout (16x16 f32): VGPR r, lane l -> element [r + 8*(l>>4)][l&15]
// ---------------------------------------------------------------------------
__global__ __launch_bounds__(32) void gemm_wmma_kernel(const float* __restrict__ X,
                                                       const float* __restrict__ W,
                                                       float* __restrict__ H) {
  const int rowBase = blockIdx.x << 4;
  const int colBase = blockIdx.y << 4;
  const int lane = threadIdx.x;
  const int half = lane >> 4;   // 0 or 1
  const int l16  = lane & 15;

  const float* xrow = X + (long long)(rowBase + l16) * D_FEAT + 2 * half;
  const float* wcol = W + (long long)(2 * half) * D_FEAT + colBase + l16;

  v8f acc = {0.f, 0.f, 0.f, 0.f, 0.f, 0.f, 0.f, 0.f};
#pragma unroll
  for (int k = 0; k < D_FEAT; k += 4) {
    v2f a = *(const v2f*)(xrow + k);            // contiguous K,K+1 pair (8B aligned)
    v2f b;
    b.x = wcol[(long long)k * D_FEAT];
    b.y = wcol[(long long)(k + 1) * D_FEAT];
    // 8 args: (neg_a, A, neg_b, B, c_mod, C, reuse_a, reuse_b)
    acc = __builtin_amdgcn_wmma_f32_16x16x4_f32(false, a, false, b, (short)0, acc,
                                                false, false);
  }

  float* hp = H + (long long)(rowBase + (half << 3)) * D_FEAT + colBase + l16;
#pragma unroll
  for (int r = 0; r < 8; ++r)
    hp[(long long)r * D_FEAT] = acc[r];
}

// ---------------------------------------------------------------------------
// agg[dst[e]][f] += h[src[e]][f] * norm[e]    (thread per (edge, feature))
// ---------------------------------------------------------------------------
__global__ __launch_bounds__(256) void scatter_kernel(const float* __restrict__ h,
                                                      const int* __restrict__ src,
                                                      const int* __restrict__ dst,
                                                      const float* __restrict__ norm,
                                                      float* __restrict__ agg,
                                                      long long total) {
  long long idx = (long long)blockIdx.x * blockDim.x + threadIdx.x;
  if (idx >= total) return;
  int ed = (int)(idx >> 7);
  int f  = (int)(idx & (D_FEAT - 1));
  float v = h[(long long)src[ed] * D_FEAT + f] * norm[ed];
  atomicAdd(&agg[(long long)dst[ed] * D_FEAT + f], v);
}

// ---------------------------------------------------------------------------
// In-place epilogue: agg = relu(agg + dinv[i]^2 * h + bias[f])
// (self-loop edge has weight 1 and norm dinv[i]*dinv[i])
// ---------------------------------------------------------------------------
__global__ __launch_bounds__(256) void finalize_kernel(float* __restrict__ agg,
                                                       const float* __restrict__ h,
                                                       const float* __restrict__ dinv,
                                                       const float* __restrict__ bias,
                                                       long long total) {
  long long idx = (long long)blockIdx.x * blockDim.x + threadIdx.x;
  if (idx >= total) return;
  int i = (int)(idx >> 7);
  int f = (int)(idx & (D_FEAT - 1));
  float d = dinv[i];
  float v = agg[idx] + d * d * h[idx] + bias[f];
  agg[idx] = v > 0.f ? v : 0.f;
}

// ---------------------------------------------------------------------------
// out[n][c] = h[n][:] . Wl[:][c] + bl[c]      (128x4 classifier, tiny)
// ---------------------------------------------------------------------------
__global__ __launch_bounds__(256) void classify_kernel(const float* __restrict__ h,
                                                       const float* __restrict__ Wl,
                                                       const float* __restrict__ bl,
                                                       float* __restrict__ out, int n) {
  int idx = blockIdx.x * blockDim.x + threadIdx.x;
  if (idx >= n * N_CLASSES) return;
  int i = idx >> 2;
  int c = idx & (N_CLASSES - 1);
  const float* hr = h + (long long)i * D_FEAT;
  float acc = bl[c];
#pragma unroll 8
  for (int k = 0; k < D_FEAT; ++k) acc += hr[k] * Wl[k * N_CLASSES + c];
  out[idx] = acc;
}

// ---------------------------------------------------------------------------
extern "C" void kernel_launch(void* const* d_in, const int* in_sizes, int n_in,
                              void* d_out, int out_size, void* d_ws, size_t ws_size,
                              hipStream_t stream) {
  const float* x  = (const float*)d_in[0];
  const int*   ei = (const int*)d_in[1];
  const float* ew = (const float*)d_in[2];
  const float* W1 = (const float*)d_in[3];
  const float* b1 = (const float*)d_in[4];
  const float* W2 = (const float*)d_in[5];
  const float* b2 = (const float*)d_in[6];
  const float* W3 = (const float*)d_in[7];
  const float* b3 = (const float*)d_in[8];
  const float* Wl = (const float*)d_in[9];
  const float* bl = (const float*)d_in[10];
  float* out = (float*)d_out;

  const int n = in_sizes[0] / D_FEAT;   // 50000 (divisible by 16)
  const int e = in_sizes[2];            // 800000
  const int* src = ei;
  const int* dst = ei + e;
  const long long nf = (long long)n * D_FEAT;

  // Workspace partition (all 16B-aligned multiples)
  float* dinv = (float*)d_ws;           // n floats (holds deg, then dinv)
  float* nrm  = dinv + n;               // e floats
  float* hbuf = nrm + e;                // n*128
  float* bufP = hbuf + nf;              // n*128
  float* bufQ = bufP + nf;              // n*128

  auto cdiv = [](long long a, long long b) { return (unsigned)((a + b - 1) / b); };

  // ---- normalization precompute ----
  zero_kernel<<<cdiv(n / 4, 256), 256, 0, stream>>>((float4*)dinv, n / 4);
  deg_accum_kernel<<<cdiv(e, 256), 256, 0, stream>>>(dst, ew, dinv, e);
  dinv_kernel<<<cdiv(n, 256), 256, 0, stream>>>(dinv, n);
  norm_kernel<<<cdiv(e, 256), 256, 0, stream>>>(src, dst, ew, dinv, nrm, e);

  dim3 ggrid(n / 16, D_FEAT / 16);
  const long long esc = (long long)e * D_FEAT;
  const unsigned sblk  = cdiv(esc, 256);
  const unsigned zblk  = cdiv(nf / 4, 256);
  const unsigned fblk  = cdiv(nf, 256);

  // ---- layer 1: x -> bufP ----
  gemm_wmma_kernel<<<ggrid, 32, 0, stream>>>(x, W1, hbuf);
  zero_kernel<<<zblk, 256, 0, stream>>>((float4*)bufP, nf / 4);
  scatter_kernel<<<sblk, 256, 0, stream>>>(hbuf, src, dst, nrm, bufP, esc);
  finalize_kernel<<<fblk, 256, 0, stream>>>(bufP, hbuf, dinv, b1, nf);

  // ---- layer 2: bufP -> bufQ ----
  gemm_wmma_kernel<<<ggrid, 32, 0, stream>>>(bufP, W2, hbuf);
  zero_kernel<<<zblk, 256, 0, stream>>>((float4*)bufQ, nf / 4);
  scatter_kernel<<<sblk, 256, 0, stream>>>(hbuf, src, dst, nrm, bufQ, esc);
  finalize_kernel<<<fblk, 256, 0, stream>>>(bufQ, hbuf, dinv, b2, nf);

  // ---- layer 3: bufQ -> bufP ----
  gemm_wmma_kernel<<<ggrid, 32, 0, stream>>>(bufQ, W3, hbuf);
  zero_kernel<<<zblk, 256, 0, stream>>>((float4*)bufP, nf / 4);
  scatter_kernel<<<sblk, 256, 0, stream>>>(hbuf, src, dst, nrm, bufP, esc);
  finalize_kernel<<<fblk, 256, 0, stream>>>(bufP, hbuf, dinv, b3, nf);

  // ---- classifier ----
  classify_kernel<<<cdiv((long long)n * N_CLASSES, 256), 256, 0, stream>>>(bufP, Wl, bl, out, n);
}